// AttentionGNNLayer_80204219285966
// MI455X (gfx1250) — compile-verified
//
#include <hip/hip_runtime.h>
#include <hip/hip_bf16.h>

typedef __attribute__((ext_vector_type(2))) float v2f;
typedef __attribute__((ext_vector_type(8))) float v8f;

// ---------------------------------------------------------------------------
// Main edge kernel: 256 threads = 8 waves; each wave owns 16 edges.
// A-matrix rows = edges (M=16), B = transposed weights (N=16 per tile),
// fp32 WMMA 16x16x4 stepped over K.
// ---------------------------------------------------------------------------
__global__ __launch_bounds__(256) void gnn_edge_wmma(
    const float* __restrict__ h, const float* __restrict__ coup,
    const float* __restrict__ W1, const float* __restrict__ b1,
    const float* __restrict__ Wq, const float* __restrict__ bq,
    const float* __restrict__ Wk, const float* __restrict__ bk,
    const int* __restrict__ send, const int* __restrict__ recv,
    float* __restrict__ out, int eHalf)
{
    __shared__ float sW1t[32 * 68];      // W1 transposed: [n][k], k padded 65->68
    __shared__ float sWqt[32 * 36];      // Wq transposed: [n][k], stride 36
    __shared__ float sWkt[32 * 36];
    __shared__ float sEF [8 * 16 * 68];  // per wave: 16 edge rows x 68 (rf|sf|c|pad)
    __shared__ int   sIdx[8 * 32];       // per wave: [0..15]=recv, [16..31]=send

    const int tid  = threadIdx.x;
    const int lane = tid & 31;
    const int wave = tid >> 5;
    const int hi   = lane >> 4;          // 0: lanes 0-15, 1: lanes 16-31
    const int lm   = lane & 15;
    const int base = (blockIdx.x * 8 + wave) * 16;   // first edge of this wave

    // ---- stage transposed weights (block-cooperative) + edge indices -------
    for (int idx = tid; idx < 32 * 68; idx += 256) {
        int n = idx / 68, kk = idx - n * 68;
        sW1t[idx] = (kk < 65) ? W1[kk * 32 + n] : 0.0f;
    }
    for (int idx = tid; idx < 32 * 36; idx += 256) {
        int n = idx / 36, kk = idx - n * 36;
        float vq = (kk < 32) ? Wq[kk * 32 + n] : 0.0f;
        float vk = (kk < 32) ? Wk[kk * 32 + n] : 0.0f;
        sWqt[idx] = vq;
        sWkt[idx] = vk;
    }
    {
        int v = (lane < 16) ? recv[base + lane] : send[base + lane - 16];
        sIdx[wave * 32 + lane] = v;
    }
    __syncthreads();

    // ---- gather rf|sf rows into LDS (16 edges x 64 floats, float4 moves) ---
    float* efW = sEF + wave * 16 * 68;
    #pragma unroll
    for (int it = 0; it < 8; ++it) {
        int s  = it * 32 + lane;          // 0..255 float4 slots
        int e  = s >> 4;                  // edge 0..15
        int q4 = s & 15;                  // slot within row: 0-7 rf, 8-15 sf
        int node = sIdx[wave * 32 + e + ((q4 < 8) ? 0 : 16)];
        int col  = (q4 & 7) * 4;
        float4 v = *(const float4*)(h + (size_t)node * 32 + col);
        *(float4*)(efW + e * 68 + q4 * 4) = v;
    }
    {   // coupling feature (k=64) + zero pad k=65..67
        int e = lm;
        if (lane < 16) {
            int ge = base + e;
            efW[e * 68 + 64] = coup[(ge < eHalf) ? ge : ge - eHalf];
            efW[e * 68 + 66] = 0.0f;
        } else {
            efW[e * 68 + 65] = 0.0f;
            efW[e * 68 + 67] = 0.0f;
        }
    }
    __syncthreads();

    // ---- WMMA: messages = ef @ W1 ; q = sf @ Wq ; k = rf @ Wk --------------
    const float* efRow = efW + lm * 68;   // A row for M=lm (both lane halves)
    const int ko = 2 * hi;                // K sub-offset per lane half

    v8f accM0 = {}, accM1 = {};
    #pragma unroll
    for (int k0 = 0; k0 < 68; k0 += 4) {
        v2f a  = *(const v2f*)(efRow + k0 + ko);
        v2f bA = *(const v2f*)(sW1t + (lm     ) * 68 + k0 + ko);
        v2f bB = *(const v2f*)(sW1t + (lm + 16) * 68 + k0 + ko);
        accM0 = __builtin_amdgcn_wmma_f32_16x16x4_f32(false, a, false, bA,
                                                      (short)0, accM0, false, false);
        accM1 = __builtin_amdgcn_wmma_f32_16x16x4_f32(false, a, false, bB,
                                                      (short)0, accM1, false, false);
    }

    v8f accQ0 = {}, accQ1 = {}, accK0 = {}, accK1 = {};
    #pragma unroll
    for (int k0 = 0; k0 < 32; k0 += 4) {
        v2f aQ  = *(const v2f*)(efRow + 32 + k0 + ko);   // sf
        v2f aK  = *(const v2f*)(efRow + k0 + ko);        // rf
        v2f bq0 = *(const v2f*)(sWqt + (lm     ) * 36 + k0 + ko);
        v2f bq1 = *(const v2f*)(sWqt + (lm + 16) * 36 + k0 + ko);
        v2f bk0 = *(const v2f*)(sWkt + (lm     ) * 36 + k0 + ko);
        v2f bk1 = *(const v2f*)(sWkt + (lm + 16) * 36 + k0 + ko);
        accQ0 = __builtin_amdgcn_wmma_f32_16x16x4_f32(false, aQ, false, bq0,
                                                      (short)0, accQ0, false, false);
        accQ1 = __builtin_amdgcn_wmma_f32_16x16x4_f32(false, aQ, false, bq1,
                                                      (short)0, accQ1, false, false);
        accK0 = __builtin_amdgcn_wmma_f32_16x16x4_f32(false, aK, false, bk0,
                                                      (short)0, accK0, false, false);
        accK1 = __builtin_amdgcn_wmma_f32_16x16x4_f32(false, aK, false, bk1,
                                                      (short)0, accK1, false, false);
    }

    // ---- attention gate: a[m] = sum_n q[m][n]*k[m][n], sigmoid ------------
    // C/D layout: VGPR j, lanes 0-15 -> edge m=j ; lanes 16-31 -> m=j+8 ;
    // N = lm + 16*tile. Reduction over N = shfl_xor within each 16-lane half.
    const float bqv0 = bq[lm], bqv1 = bq[lm + 16];
    const float bkv0 = bk[lm], bkv1 = bk[lm + 16];
    float gate[8];
    #pragma unroll
    for (int j = 0; j < 8; ++j) {
        float p = (accQ0[j] + bqv0) * (accK0[j] + bkv0)
                + (accQ1[j] + bqv1) * (accK1[j] + bkv1);
        p += __shfl_xor(p, 1, 32);
        p += __shfl_xor(p, 2, 32);
        p += __shfl_xor(p, 4, 32);
        p += __shfl_xor(p, 8, 32);
        gate[j] = 1.0f / (1.0f + __expf(-p));   // sigmoid(a[j + 8*hi])
    }

    // ---- relu(msg + b1) * gate, scatter-add to out[recv] -------------------
    const float b10 = b1[lm], b11 = b1[lm + 16];
    const int rbase = wave * 32 + 8 * hi;
    #pragma unroll
    for (int j = 0; j < 8; ++j) {
        int r = sIdx[rbase + j];               // receiver of edge j + 8*hi
        float m0 = fmaxf(accM0[j] + b10, 0.0f) * gate[j];
        float m1 = fmaxf(accM1[j] + b11, 0.0f) * gate[j];
        atomicAdd(out + (size_t)r * 32 + lm,      m0);
        atomicAdd(out + (size_t)r * 32 + lm + 16, m1);
    }
}

// ---------------------------------------------------------------------------
// Scalar tail kernel (only used if nEdges % 128 != 0; zero edges here).
// ---------------------------------------------------------------------------
__global__ void gnn_edge_tail(
    const float* __restrict__ h, const float* __restrict__ coup,
    const float* __restrict__ W1, const float* __restrict__ b1,
    const float* __restrict__ Wq, const float* __restrict__ bq,
    const float* __restrict__ Wk, const float* __restrict__ bk,
    const int* __restrict__ send, const int* __restrict__ recv,
    float* __restrict__ out, int start, int nEdges, int eHalf)
{
    int e = start + blockIdx.x * blockDim.x + threadIdx.x;
    if (e >= nEdges) return;
    int r = recv[e], s = send[e];
    float c = coup[(e < eHalf) ? e : e - eHalf];
    const float* rf = h + (size_t)r * 32;
    const float* sf = h + (size_t)s * 32;
    float a = 0.0f;
    for (int n = 0; n < 32; ++n) {
        float qn = bq[n], kn = bk[n];
        for (int d = 0; d < 32; ++d) {
            qn += sf[d] * Wq[d * 32 + n];
            kn += rf[d] * Wk[d * 32 + n];
        }
        a += qn * kn;
    }
    float sig = 1.0f / (1.0f + __expf(-a));
    for (int n = 0; n < 32; ++n) {
        float m = b1[n];
        for (int d = 0; d < 32; ++d)
            m += rf[d] * W1[d * 32 + n] + sf[d] * W1[(32 + d) * 32 + n];
        m += c * W1[64 * 32 + n];
        atomicAdd(out + (size_t)r * 32 + n, fmaxf(m, 0.0f) * sig);
    }
}

__global__ void gnn_relu_inplace(float* __restrict__ out, int n)
{
    int i = blockIdx.x * blockDim.x + threadIdx.x;
    if (i < n) out[i] = fmaxf(out[i], 0.0f);
}

// ---------------------------------------------------------------------------
extern "C" void kernel_launch(void* const* d_in, const int* in_sizes, int n_in,
                              void* d_out, int out_size, void* d_ws, size_t ws_size,
                              hipStream_t stream)
{
    const float* h    = (const float*)d_in[0];
    const float* coup = (const float*)d_in[1];
    const float* W1   = (const float*)d_in[2];
    const float* b1   = (const float*)d_in[3];
    const float* Wq   = (const float*)d_in[4];
    const float* bq   = (const float*)d_in[5];
    const float* Wk   = (const float*)d_in[6];
    const float* bk   = (const float*)d_in[7];
    const int*   send = (const int*)d_in[8];
    const int*   recv = (const int*)d_in[9];
    float* out = (float*)d_out;

    const int eHalf  = in_sizes[1];
    const int nEdges = in_sizes[8];

    // zero the accumulator (replay-safe: every launch re-zeros)
    hipMemsetAsync(d_out, 0, (size_t)out_size * sizeof(float), stream);

    const int edgesPerBlock = 128;                 // 8 waves x 16 edges
    const int mainBlocks = nEdges / edgesPerBlock;
    if (mainBlocks > 0) {
        gnn_edge_wmma<<<mainBlocks, 256, 0, stream>>>(
            h, coup, W1, b1, Wq, bq, Wk, bk, send, recv, out, eHalf);
    }
    const int done = mainBlocks * edgesPerBlock;
    const int rem  = nEdges - done;
    if (rem > 0) {
        gnn_edge_tail<<<(rem + 63) / 64, 64, 0, stream>>>(
            h, coup, W1, b1, Wq, bq, Wk, bk, send, recv, out, done, nEdges, eHalf);
    }
    gnn_relu_inplace<<<(out_size + 255) / 256, 256, 0, stream>>>(out, out_size);
}